// SHLPTHead_1924145349120
// MI455X (gfx1250) — compile-verified
//
#include <hip/hip_runtime.h>
#include <hip/hip_bf16.h>

typedef __attribute__((ext_vector_type(2))) float v2f;
typedef __attribute__((ext_vector_type(8))) float v8f;

#define B_   32
#define S_   512
#define P_   16
#define L_   100
#define D_   1024
#define BOT_ 100
#define BOTP 112          // BOT padded to multiple of 16
#define NS_  8            // S-chunks for max reduction
#define SCH_ (S_ / NS_)   // 64
#define LD_  (L_ * D_)    // 102400

// ---------------------------------------------------------------------------
// Stage 0: zero-pad W_down[100,1024] -> Wdp[112,1024] (rows 100..111 = 0)
// Removes all boundary conditionals from the WMMA inner loops.
// ---------------------------------------------------------------------------
__global__ void kpad_wd(const float* __restrict__ Wd, float* __restrict__ Wdp) {
    int row = blockIdx.x;              // 0..111
    int d0 = threadIdx.x * 4;          // 256 threads * 4 = 1024
    float4 v = {0.f, 0.f, 0.f, 0.f};
    if (row < BOT_) v = *(const float4*)(Wd + (size_t)row * D_ + d0);
    *(float4*)(Wdp + (size_t)row * D_ + d0) = v;
}

// ---------------------------------------------------------------------------
// Stage 1a: partial max over S chunks: c_embed[32,512,1024] -> part[32,8,1024]
// ---------------------------------------------------------------------------
__global__ void kmax_partial(const float* __restrict__ c, float* __restrict__ part) {
    int blk = blockIdx.x;          // b*NS_ + chunk
    int b  = blk / NS_;
    int ch = blk % NS_;
    int d0 = threadIdx.x * 4;      // 256 threads * 4 = 1024
    const float* base = c + ((size_t)b * S_ + (size_t)ch * SCH_) * D_ + d0;
    float4 m = *(const float4*)base;
    for (int s = 1; s < SCH_; ++s) {
        if (s + 8 < SCH_) __builtin_prefetch(base + (size_t)(s + 8) * D_, 0, 1);
        float4 v = *(const float4*)(base + (size_t)s * D_);
        m.x = fmaxf(m.x, v.x); m.y = fmaxf(m.y, v.y);
        m.z = fmaxf(m.z, v.z); m.w = fmaxf(m.w, v.w);
    }
    *(float4*)(part + ((size_t)b * NS_ + ch) * D_ + d0) = m;
}

// Stage 1b: reduce the 8 partials -> avg_c[32,1024]
__global__ void kmax_finish(const float* __restrict__ part, float* __restrict__ avg_c) {
    int b = blockIdx.x;
    int d0 = threadIdx.x * 4;
    const float* base = part + (size_t)b * NS_ * D_ + d0;
    float4 m = *(const float4*)base;
    for (int ch = 1; ch < NS_; ++ch) {
        float4 v = *(const float4*)(base + (size_t)ch * D_);
        m.x = fmaxf(m.x, v.x); m.y = fmaxf(m.y, v.y);
        m.z = fmaxf(m.z, v.z); m.w = fmaxf(m.w, v.w);
    }
    *(float4*)(avg_c + (size_t)b * D_ + d0) = m;
}

// Stage 1c: max over L: previous_prompts[16,100,1024] -> avg_pre[16,1024]
__global__ void kmax_pre(const float* __restrict__ prev, float* __restrict__ avg_pre) {
    int p = blockIdx.x;
    int d0 = threadIdx.x * 4;
    const float* base = prev + (size_t)p * L_ * D_ + d0;
    float4 m = *(const float4*)base;
    for (int l = 1; l < L_; ++l) {
        float4 v = *(const float4*)(base + (size_t)l * D_);
        m.x = fmaxf(m.x, v.x); m.y = fmaxf(m.y, v.y);
        m.z = fmaxf(m.z, v.z); m.w = fmaxf(m.w, v.w);
    }
    *(float4*)(avg_pre + (size_t)p * D_ + d0) = m;
}

// ---------------------------------------------------------------------------
// Stage 2: Hs = SiLU(avg_c @ Wdp.T)  via V_WMMA_F32_16X16X4_F32
// avg_c[32,1024], Wdp[112,1024] -> Hs[32,112]. One wave per 16x16 tile.
// Grid = 2 (M tiles) * 7 (N tiles) = 14 blocks of 32. Branch-free inner loop.
// ---------------------------------------------------------------------------
__global__ void kgemm_down_silu(const float* __restrict__ avg_c,
                                const float* __restrict__ Wdp,
                                float* __restrict__ Hs) {
    int tile = blockIdx.x;
    int mT = tile / 7, nT = tile % 7;
    int lane = threadIdx.x;            // 0..31, EXEC all ones
    int half = lane >> 4, lid = lane & 15;
    int m = mT * 16 + lid;             // A row (always < 32)
    int n = nT * 16 + lid;             // B col (padded rows are zero)
    const float* arow = avg_c + (size_t)m * D_ + half * 2;
    const float* brow = Wdp   + (size_t)n * D_ + half * 2;
    v8f acc = {0.f, 0.f, 0.f, 0.f, 0.f, 0.f, 0.f, 0.f};
#pragma unroll 4
    for (int k = 0; k < D_; k += 4) {
        v2f a  = *(const v2f*)(arow + k);
        v2f bm = *(const v2f*)(brow + k);
        acc = __builtin_amdgcn_wmma_f32_16x16x4_f32(false, a, false, bm,
                                                    (short)0, acc, false, false);
    }
    for (int r = 0; r < 8; ++r) {
        int mm = mT * 16 + half * 8 + r;
        float v = acc[r];
        // SiLU fused into store; fast rcp instead of IEEE divide chain
        Hs[mm * BOTP + n] = v * __builtin_amdgcn_rcpf(1.f + __expf(-v));
    }
}

// ---------------------------------------------------------------------------
// Stage 3: X = Hs @ W_up.T ; Hs[32,100(pad112)], W_up[1024,100] -> X[32,1024]
// Grid = 2 * 64 = 128 blocks of 32.
// ---------------------------------------------------------------------------
__global__ void kgemm_up(const float* __restrict__ Hs,
                         const float* __restrict__ Wu,
                         float* __restrict__ X) {
    int tile = blockIdx.x;
    int mT = tile / 64, nT = tile % 64;
    int lane = threadIdx.x;
    int half = lane >> 4, lid = lane & 15;
    int m = mT * 16 + lid;
    int n = nT * 16 + lid;             // 0..1023, always valid
    const float* arow = Hs + (size_t)m * BOTP + half * 2;
    const float* brow = Wu + (size_t)n * BOT_ + half * 2;
    v8f acc = {0.f, 0.f, 0.f, 0.f, 0.f, 0.f, 0.f, 0.f};
#pragma unroll 5
    for (int k = 0; k < BOT_; k += 4) {   // 25 iterations
        v2f a  = *(const v2f*)(arow + k);
        v2f bm = *(const v2f*)(brow + k);
        acc = __builtin_amdgcn_wmma_f32_16x16x4_f32(false, a, false, bm,
                                                    (short)0, acc, false, false);
    }
    for (int r = 0; r < 8; ++r) {
        int mm = mT * 16 + half * 8 + r;
        X[mm * D_ + n] = acc[r];
    }
}

// ---------------------------------------------------------------------------
// Stage 4: LayerNorm rows of X[32,1024] -> Xn (one block per row)
// ---------------------------------------------------------------------------
__global__ void kln(const float* __restrict__ X,
                    const float* __restrict__ gamma,
                    const float* __restrict__ beta,
                    float* __restrict__ Xn) {
    __shared__ float ssum[256];
    __shared__ float ssq[256];
    int b = blockIdx.x, t = threadIdx.x;
    float s = 0.f, q = 0.f;
    for (int i = t; i < D_; i += 256) {
        float x = X[b * D_ + i];
        s += x; q += x * x;
    }
    ssum[t] = s; ssq[t] = q;
    __syncthreads();
    for (int off = 128; off > 0; off >>= 1) {
        if (t < off) { ssum[t] += ssum[t + off]; ssq[t] += ssq[t + off]; }
        __syncthreads();
    }
    float mu  = ssum[0] * (1.f / D_);
    float var = ssq[0] * (1.f / D_) - mu * mu;
    float inv = rsqrtf(var + 1e-5f);
    for (int i = t; i < D_; i += 256) {
        float x = X[b * D_ + i];
        Xn[b * D_ + i] = (x - mu) * inv * gamma[i] + beta[i];
    }
}

// ---------------------------------------------------------------------------
// Stage 5: scores[32,16] = Xn @ avg_pre.T (P=16 == exactly one N tile)
// Grid = 2 blocks of 32 (one per M tile).
// ---------------------------------------------------------------------------
__global__ void kattn(const float* __restrict__ Xn,
                      const float* __restrict__ avg_pre,
                      float* __restrict__ scores) {
    int mT = blockIdx.x;
    int lane = threadIdx.x;
    int half = lane >> 4, lid = lane & 15;
    int m = mT * 16 + lid;
    const float* arow = Xn      + (size_t)m   * D_ + half * 2;
    const float* brow = avg_pre + (size_t)lid * D_ + half * 2;
    v8f acc = {0.f, 0.f, 0.f, 0.f, 0.f, 0.f, 0.f, 0.f};
#pragma unroll 4
    for (int k = 0; k < D_; k += 4) {
        v2f a  = *(const v2f*)(arow + k);
        v2f bm = *(const v2f*)(brow + k);
        acc = __builtin_amdgcn_wmma_f32_16x16x4_f32(false, a, false, bm,
                                                    (short)0, acc, false, false);
    }
    for (int r = 0; r < 8; ++r) {
        int mm = mT * 16 + half * 8 + r;
        scores[mm * P_ + lid] = acc[r];   // TEMP == 1.0
    }
}

// ---------------------------------------------------------------------------
// Stage 6: softmax -> threshold(0.05) -> renormalize. 32 rows, thread = row.
// ---------------------------------------------------------------------------
__global__ void kfilt(const float* __restrict__ scores, float* __restrict__ filt) {
    int b = threadIdx.x;               // 32 threads, 1 block
    float v[P_];
    float mx = -3.4e38f;
    for (int p = 0; p < P_; ++p) { v[p] = scores[b * P_ + p]; mx = fmaxf(mx, v[p]); }
    float sum = 0.f;
    for (int p = 0; p < P_; ++p) { v[p] = __expf(v[p] - mx); sum += v[p]; }
    float inv = __builtin_amdgcn_rcpf(sum);
    float s2 = 0.f;
    for (int p = 0; p < P_; ++p) {
        float nv = v[p] * inv;
        nv = (nv > 0.05f) ? nv : 0.f;
        v[p] = nv; s2 += nv;
    }
    float inv2 = __builtin_amdgcn_rcpf(s2);
    for (int p = 0; p < P_; ++p) filt[b * P_ + p] = v[p] * inv2;
}

// ---------------------------------------------------------------------------
// Stage 7: out = filt[32,16] x prev[16,102400] + prompt_embed  (WMMA, C = pe)
// M tiles = 2, N tiles = 6400 -> 12800 waves -> 1600 blocks of 256.
// Residual add fused via the WMMA C operand.
// ---------------------------------------------------------------------------
__global__ void kcombine(const float* __restrict__ filt,
                         const float* __restrict__ prev,
                         const float* __restrict__ pe,
                         float* __restrict__ out) {
    int wid  = (blockIdx.x * blockDim.x + threadIdx.x) >> 5;  // global wave id
    int lane = threadIdx.x & 31;
    int mT = wid & 1;
    int nT = wid >> 1;                 // 0..6399
    int half = lane >> 4, lid = lane & 15;
    int m = mT * 16 + lid;
    size_t ncol = (size_t)nT * 16 + lid;   // flattened (l,d) column

    v8f acc;                               // C := prompt_embed tile (fused +)
#pragma unroll
    for (int r = 0; r < 8; ++r)
        acc[r] = pe[(size_t)(mT * 16 + half * 8 + r) * LD_ + ncol];

#pragma unroll
    for (int k = 0; k < P_; k += 4) {
        int k0 = k + half * 2;
        v2f a = *(const v2f*)(filt + m * P_ + k0);   // contiguous pair
        v2f bm;
        bm.x = prev[(size_t)k0 * LD_ + ncol];
        bm.y = prev[(size_t)(k0 + 1) * LD_ + ncol];
        acc = __builtin_amdgcn_wmma_f32_16x16x4_f32(false, a, false, bm,
                                                    (short)0, acc, false, false);
    }
#pragma unroll
    for (int r = 0; r < 8; ++r)
        out[(size_t)(mT * 16 + half * 8 + r) * LD_ + ncol] = acc[r];
}

// ---------------------------------------------------------------------------
extern "C" void kernel_launch(void* const* d_in, const int* in_sizes, int n_in,
                              void* d_out, int out_size, void* d_ws, size_t ws_size,
                              hipStream_t stream) {
    const float* c_embed = (const float*)d_in[0];   // [32,512,1024]
    const float* prev    = (const float*)d_in[1];   // [16,100,1024]
    const float* pe      = (const float*)d_in[2];   // [32,100,1024]
    const float* Wd      = (const float*)d_in[3];   // [100,1024]
    const float* Wu      = (const float*)d_in[4];   // [1024,100]
    const float* gamma   = (const float*)d_in[5];   // [1024]
    const float* beta    = (const float*)d_in[6];   // [1024]

    float* out      = (float*)d_out;                      // [32*100*1024]
    float* filt_out = out + (size_t)B_ * LD_;             // [32*16] tail of d_out

    float* ws      = (float*)d_ws;
    float* part    = ws;                       // 32*8*1024  = 262144
    float* avg_c   = part    + B_ * NS_ * D_;  // 32768
    float* avg_pre = avg_c   + B_ * D_;        // 16384
    float* Hs      = avg_pre + P_ * D_;        // 32*112 = 3584
    float* X       = Hs      + B_ * BOTP;      // 32768
    float* Xn      = X       + B_ * D_;        // 32768
    float* scores  = Xn      + B_ * D_;        // 512
    float* Wdp     = scores  + B_ * P_;        // 112*1024 = 114688

    kpad_wd        <<<BOTP,     256, 0, stream>>>(Wd, Wdp);
    kmax_partial   <<<B_ * NS_, 256, 0, stream>>>(c_embed, part);
    kmax_finish    <<<B_,       256, 0, stream>>>(part, avg_c);
    kmax_pre       <<<P_,       256, 0, stream>>>(prev, avg_pre);
    kgemm_down_silu<<<14,        32, 0, stream>>>(avg_c, Wdp, Hs);
    kgemm_up       <<<128,       32, 0, stream>>>(Hs, Wu, X);
    kln            <<<B_,       256, 0, stream>>>(X, gamma, beta, Xn);
    kattn          <<<2,         32, 0, stream>>>(Xn, avg_pre, scores);
    kfilt          <<<1,         32, 0, stream>>>(scores, filt_out);
    kcombine       <<<1600,     256, 0, stream>>>(filt_out, prev, pe, out);
}